// ChirpLinker_84662395338807
// MI455X (gfx1250) — compile-verified
//
#include <hip/hip_runtime.h>

#define W_DIM 128
#define K_DIM 32
#define B_DIM 16
#define NT 256
#define NCELL (W_DIM * K_DIM) /* 4096 */

#define MAX_DF_C 0.05f
#define MAX_DPHI_C 0.5f
#define MAX_DA_C 0.5f
#define MIN_LENGTH_C 2
#define PI_F 3.14159265358979323846f
#define TWO_PI_F 6.28318530717958647692f
#define NEG_INF (-__builtin_inff())

typedef float v2f __attribute__((ext_vector_type(2)));
typedef float v8f __attribute__((ext_vector_type(8)));

__device__ __forceinline__ float wrapf(float x) {
  // JAX: (x + pi) % (2pi) - pi  with mod sign following the (positive) divisor
  float y = fmodf(x + PI_F, TWO_PI_F);
  y = (y < 0.0f) ? (y + TWO_PI_F) : y;
  return y - PI_F;
}

__device__ __forceinline__ unsigned fkey(float f) {
  // monotonic float -> uint mapping (handles -inf correctly)
  unsigned u = __float_as_uint(f);
  return (u & 0x80000000u) ? ~u : (u | 0x80000000u);
}

// One workgroup per batch element. All state lives in (dynamic) LDS.
__global__ __launch_bounds__(NT) void chirp_link_kernel(
    const float* __restrict__ tokens, float* __restrict__ out,
    int* __restrict__ counts) {
  extern __shared__ unsigned char smem[];
  // layout (96 KB total):
  unsigned long long* keys = (unsigned long long*)smem;        // [4096] 32KB (phase >=3)
  unsigned* Emask = (unsigned*)smem;                           // alias: first 16KB (phase 1-2)
  float* cand = (float*)(smem + 16384);                        // alias: 4KB  (phase 2)
  float* snrL = (float*)(smem + 32768);                        // [4096] 16KB
  float* bestL = (float*)(smem + 49152);                       // [4096] 16KB
  float* chain2 = (float*)(smem + 65536);                      // [2048]  8KB
  short* memberL = (short*)(smem + 73728);                     // [4096]  8KB
  signed char* predL = (signed char*)(smem + 81920);           // [4096]  4KB
  unsigned char* LL = (unsigned char*)(smem + 86016);          // [4096]  4KB
  unsigned char* usedL = (unsigned char*)(smem + 90112);       // [4096]  4KB
  signed char* succL = (signed char*)(smem + 94208);           // [4096]  4KB

  const int b = blockIdx.x;
  const int tid = threadIdx.x;
  const float* tok = tokens + (size_t)b * NCELL * 9;
#define TOK(w, k, c) tok[(((w)*K_DIM) + (k)) * 9 + (c)]

  // CDNA5 prefetch path (global_prefetch_b8): pull this batch's tokens toward L2
  __builtin_prefetch(tok + (size_t)tid * 144, 0, 0);

  // ---- P0: load snr, init state ----
  for (int i = tid; i < NCELL; i += NT) {
    snrL[i] = tok[i * 9 + 0];
    memberL[i] = -1;
    predL[i] = -1;
    succL[i] = -1;
    usedL[i] = 0;
    LL[i] = 0;
  }
  for (int i = tid; i < NCELL / 2; i += NT) chain2[i] = 0.0f;
  __syncthreads();

  // ---- P1: edge bitmasks Emask[w*K + k] bit j  <=>  edge (w,j)->(w+1,k) ----
  for (int idx = tid; idx < (W_DIM - 1) * K_DIM; idx += NT) {
    int w = idx >> 5, k = idx & 31;
    unsigned m = 0;
    float snrn = snrL[(w + 1) * K_DIM + k];
    if (snrn > 0.0f) {
      float fs = TOK(w + 1, k, 3);
      float As = TOK(w + 1, k, 5);
      float ps = TOK(w + 1, k, 7);
      for (int j = 0; j < K_DIM; ++j) {
        if (!(snrL[w * K_DIM + j] > 0.0f)) continue;
        float fe = TOK(w, j, 4), ae = TOK(w, j, 6), pe = TOK(w, j, 8);
        float fm = 0.5f * (fe + fs);
        bool f_ok = !((fm > 0.0f) && (fabsf(fe - fs) / fm > MAX_DF_C));
        bool p_ok = fabsf(wrapf(ps - pe)) <= MAX_DPHI_C;
        float am = fmaxf(ae, As);
        bool a_ok = !((am > 0.0f) && (fabsf(ae - As) / am > MAX_DA_C));
        if (f_ok && p_ok && a_ok) m |= (1u << j);
      }
    }
    Emask[idx] = m;
  }
  __syncthreads();

  // ---- P2: max-plus DP on wave 0. Outer sum best_prev[j]+snr[w][k] computed
  //          exactly via V_WMMA_F32_16X16X4_F32 ([bp,1] x [[1],[snr]]), then
  //          masked first-index argmax per column (matches jnp argmax ties).
  if (tid < 32) {
    int lane = tid;
    float s0 = snrL[lane];
    bestL[lane] = (s0 > 0.0f) ? s0 : NEG_INF;
    bool lo = lane < 16;
    int lk = lane & 15;
    int hi8 = (lane >> 4) * 8;
    for (int w = 1; w < W_DIM; ++w) {
      const float* bprev = bestL + (w - 1) * K_DIM;
      const float* srow = snrL + w * K_DIM;
      // A (16x4 f32): lanes0-15 hold (K0,K1); lanes16-31 hold (K2,K3)=0
      v2f a0, a1, b0, b1;
      a0.x = lo ? bprev[lk] : 0.0f;
      a0.y = lo ? 1.0f : 0.0f;
      a1.x = lo ? bprev[16 + lk] : 0.0f;
      a1.y = a0.y;
      // B (4x16 f32): row K0 = ones, row K1 = snr chunk, K2/K3 = 0
      b0.x = lo ? 1.0f : 0.0f;
      b0.y = lo ? srow[lk] : 0.0f;
      b1.x = b0.x;
      b1.y = lo ? srow[16 + lk] : 0.0f;
      v8f cz = {0.0f, 0.0f, 0.0f, 0.0f, 0.0f, 0.0f, 0.0f, 0.0f};
      v8f d00 = __builtin_amdgcn_wmma_f32_16x16x4_f32(false, a0, false, b0,
                                                      (short)0, cz, false, false);
      v8f d01 = __builtin_amdgcn_wmma_f32_16x16x4_f32(false, a0, false, b1,
                                                      (short)0, cz, false, false);
      v8f d10 = __builtin_amdgcn_wmma_f32_16x16x4_f32(false, a1, false, b0,
                                                      (short)0, cz, false, false);
      v8f d11 = __builtin_amdgcn_wmma_f32_16x16x4_f32(false, a1, false, b1,
                                                      (short)0, cz, false, false);
      // D layout: elem r -> row M = r + 8*(lane/16), col N = lane%16 (lane&15)
      int col = lane & 15;
#pragma unroll
      for (int r = 0; r < 8; ++r) {
        cand[(hi8 + r) * K_DIM + col] = d00[r];
        cand[(hi8 + r) * K_DIM + 16 + col] = d01[r];
        cand[(16 + hi8 + r) * K_DIM + col] = d10[r];
        cand[(16 + hi8 + r) * K_DIM + 16 + col] = d11[r];
      }
      // per-column masked argmax (lane = column k); strict > keeps lowest j
      unsigned m = Emask[(w - 1) * K_DIM + lane];
      float maxv = NEG_INF;
      int argj = -1;
      for (int j = 0; j < K_DIM; ++j) {
        if (m & (1u << j)) {
          float v = cand[j * K_DIM + lane];  // = best_prev[j] + snr[w][lane]
          if (v > maxv) { maxv = v; argj = j; }
        }
      }
      bestL[w * K_DIM + lane] = (argj >= 0) ? maxv : NEG_INF;
      predL[w * K_DIM + lane] = (signed char)argj;
    }
  }
  __syncthreads();

  // ---- P3a: sort keys (stable descending by score, ascending index) ----
  for (int i = tid; i < NCELL; i += NT) {
    float bv = bestL[i];
    float score = (bv > NEG_INF) ? bv : ((snrL[i] > 0.0f) ? snrL[i] : NEG_INF);
    unsigned long long key =
        ((unsigned long long)fkey(score) << 12) |
        (unsigned long long)((i ^ 0xFFF) & 0xFFF);
    keys[i] = ~key;  // ascending sort of ~key == stable descending by (score, -idx)
  }
  // ---- P3b: chain-length scan on wave 0 (gather across lanes via shfl) ----
  if (tid < 32) {
    int lane = tid;
    int Lp = ((bestL[lane] > NEG_INF) || (snrL[lane] > 0.0f)) ? 1 : 0;
    LL[lane] = (unsigned char)Lp;
    for (int w = 1; w < W_DIM; ++w) {
      int p = predL[w * K_DIM + lane];
      int Lfetch = __shfl(Lp, (p >= 0) ? p : 0, 32);
      int Lc;
      if (p >= 0) {
        Lc = Lfetch + 1;
      } else {
        bool valid = (bestL[w * K_DIM + lane] > NEG_INF) ||
                     (snrL[w * K_DIM + lane] > 0.0f);
        Lc = valid ? 1 : 0;
      }
      LL[w * K_DIM + lane] = (unsigned char)Lc;
      Lp = Lc;
    }
  }
  __syncthreads();

  // ---- P4: bitonic sort (ascending on ~key) in LDS ----
  for (unsigned size = 2; size <= (unsigned)NCELL; size <<= 1) {
    for (unsigned stride = size >> 1; stride > 0; stride >>= 1) {
      for (unsigned t = tid; t < (unsigned)NCELL / 2; t += NT) {
        unsigned i = 2 * t - (t & (stride - 1));
        unsigned j = i + stride;
        unsigned long long a = keys[i], c = keys[j];
        bool asc = ((i & size) == 0);
        if (asc ? (a > c) : (a < c)) { keys[i] = c; keys[j] = a; }
      }
      __syncthreads();
    }
  }

  // ---- P5: sequential greedy chain claiming (thread 0) ----
  if (tid == 0) {
    int cnt = 0;
    unsigned char cols[W_DIM];
    for (int pos = 0; pos < NCELL; ++pos) {
      int e = (int)(keys[pos] & 0xFFFull);  // low 12 bits of ~key == index
      int Le = LL[e];
      if (Le < 1) continue;  // invalid (score == -inf)
      int w = e >> 5, k = e & 31;
      // backtrack path, checking overlap with already-claimed cells
      bool free_ = true;
      int c = k;
      for (int s = 0; s < Le; ++s) {
        int row = w - s;
        int cell = row * K_DIM + c;
        if (usedL[cell]) { free_ = false; break; }
        cols[s] = (unsigned char)c;
        c = (row > 0) ? (int)predL[cell] : -1;
      }
      if (!free_) continue;
      bool enrich = (Le >= MIN_LENGTH_C);
      int prev = -1;  // succ of head = -1; else downstream column
      for (int s = 0; s < Le; ++s) {
        int row = w - s;
        int cc = cols[s];
        int cell = row * K_DIM + cc;
        usedL[cell] = 1;  // claimed even when not enriched (L==1)
        if (enrich) {
          memberL[cell] = (short)cnt;
          succL[cell] = (signed char)prev;
        }
        prev = cc;
      }
      if (enrich) {
        float sum2 = 0.0f;  // accumulate in ascending-w order (segment_sum order)
        for (int s = Le - 1; s >= 0; --s) {
          float sv = snrL[(w - s) * K_DIM + cols[s]];
          sum2 += sv * sv;
        }
        chain2[cnt] = sum2;
        ++cnt;
      }
    }
    counts[b] = cnt;
  }
  __syncthreads();

  // ---- P6: reconstruct the 10-channel output ----
  for (int i = tid; i < NCELL; i += NT) {
    int w = i >> 5, k = i & 31;
    int m = memberL[i];
    float sn = (m >= 0) ? sqrtf(chain2[m]) : snrL[i];
    float t1 = TOK(w, k, 1), t2 = TOK(w, k, 2);
    float f_s0 = TOK(w, k, 3), f_e0 = TOK(w, k, 4);
    float A_s0 = TOK(w, k, 5), A_e0 = TOK(w, k, 6);
    float ps0 = TOK(w, k, 7), pe0 = TOK(w, k, 8);
    float o3 = f_s0, o4 = f_e0, o5 = A_s0, o6 = A_e0, o7 = ps0, o8 = pe0;
    int sc = succL[i];
    if (sc >= 0) {  // has successor at (w+1, sc): update *_e / pe
      float fsn = TOK(w + 1, sc, 3), Asn = TOK(w + 1, sc, 5),
            psn = TOK(w + 1, sc, 7);
      float corr = wrapf(psn - pe0);
      o4 = 0.5f * (f_e0 + fsn);
      o6 = 0.5f * (A_e0 + Asn);
      o8 = pe0 + 0.5f * corr;
    }
    int p = predL[i];
    if (m >= 0 && p >= 0) {  // scatter target of chain predecessor (w-1, p)
      float feS = TOK(w - 1, p, 4), AeS = TOK(w - 1, p, 6),
            peS = TOK(w - 1, p, 8);
      float corr = wrapf(ps0 - peS);
      o3 = 0.5f * (feS + f_s0);
      o5 = 0.5f * (AeS + A_s0);
      o7 = ps0 - 0.5f * corr;
    }
    float* o = out + ((size_t)b * NCELL + i) * 10;
    o[0] = sn; o[1] = t1; o[2] = t2; o[3] = o3; o[4] = o4;
    o[5] = o5; o[6] = o6; o[7] = o7; o[8] = o8;
    o[9] = (m >= 0) ? (float)m : -1.0f;  // local cid; kernel 2 adds batch offset
  }
#undef TOK
}

// Second kernel: exclusive prefix over per-batch chain counts, apply to ch9.
__global__ __launch_bounds__(NT) void chirp_offset_kernel(
    float* __restrict__ out, const int* __restrict__ counts) {
  __shared__ int offs[B_DIM];
  if (threadIdx.x == 0) {
    int acc = 0;
    for (int b = 0; b < B_DIM; ++b) {
      offs[b] = acc;
      acc += counts[b];
    }
  }
  __syncthreads();
  const int total = B_DIM * NCELL;
  for (int i = threadIdx.x; i < total; i += NT) {
    int b = i >> 12;  // i / NCELL
    float* p = out + (size_t)i * 10 + 9;
    float v = *p;
    if (v >= 0.0f) *p = v + (float)offs[b];
  }
}

extern "C" void kernel_launch(void* const* d_in, const int* in_sizes, int n_in,
                              void* d_out, int out_size, void* d_ws,
                              size_t ws_size, hipStream_t stream) {
  (void)in_sizes; (void)n_in; (void)out_size; (void)ws_size;
  const float* tokens = (const float*)d_in[0];
  float* out = (float*)d_out;
  int* counts = (int*)d_ws;  // 16 ints of scratch, written every call
  const size_t smem = 98304;  // 96 KB dynamic LDS (CDNA5 WGP: 320 KB)
  chirp_link_kernel<<<dim3(B_DIM), dim3(NT), smem, stream>>>(tokens, out, counts);
  chirp_offset_kernel<<<dim3(1), dim3(NT), 0, stream>>>(out, counts);
}